// RotateMyLayer_69844758167604
// MI455X (gfx1250) — compile-verified
//
#include <hip/hip_runtime.h>

// ---------------------------------------------------------------------------
// STN warp + pad for CDNA5 (gfx1250, wave32).
//   feature_map:        [8, 64, 512, 256] f32
//   transform_matrixs:  [10, 2, 3]        f32
//   box_masks:          [10]              i32
//   box_widths:         [10]              i32
//   out:                [10, 8, 384, 256] f32
//
// Memory-bound gather/interp (~31.5 MB out + L2-cached gathers; a few us at
// 23.3 TB/s). One wave handles one (box, out_row, 16-col tile). The affine
// einsum coords = theta x [xn; yn; 1] runs on V_WMMA_F32_16X16X4_F32; the
// bilinear sampling streams channels with coalesced B128 loads and
// non-temporal B128 stores (output is write-once).
// ---------------------------------------------------------------------------

typedef float v2f __attribute__((ext_vector_type(2)));
typedef float v4f __attribute__((ext_vector_type(4)));
typedef float v8f __attribute__((ext_vector_type(8)));

#define MAX_W 384
#define OUT_H 8
#define N_BOX 10
#define FM_H 64
#define FM_W 512
#define FM_C 256
#define TILE 16
#define TILES_PER_ROW (MAX_W / TILE)            // 24
#define N_WAVES (N_BOX * OUT_H * TILES_PER_ROW) // 1920

__device__ __forceinline__ float readlane_f32(float v, int lane) {
    return __builtin_bit_cast(
        float, __builtin_amdgcn_readlane(__builtin_bit_cast(unsigned int, v),
                                         (unsigned int)lane));
}

__device__ __forceinline__ void store_nt_f4(float* p, v4f v) {
    __builtin_nontemporal_store(v, (v4f*)p);
}

__global__ __launch_bounds__(128)
void stn_warp_wmma_kernel(const float* __restrict__ fmap,
                          const float* __restrict__ theta,
                          const int*   __restrict__ box_masks,
                          const int*   __restrict__ box_widths,
                          float*       __restrict__ out) {
    const int wave = (int)((blockIdx.x * blockDim.x + threadIdx.x) >> 5);
    const int lane = (int)(threadIdx.x & 31);
    if (wave >= N_WAVES) return;                 // wave-uniform; never taken w/ exact grid

    const int tile  = wave % TILES_PER_ROW;
    const int h     = (wave / TILES_PER_ROW) % OUT_H;
    const int n     =  wave / (TILES_PER_ROW * OUT_H);
    const int jbase = tile * TILE;

    const int   b    = box_masks[n];
    const int   bw   = box_widths[n];
    const int   wmax = bw > 2 ? bw : 2;          // jnp.maximum(box_widths, 2)
    const float wf   = (float)wmax;

    const float t00 = theta[n * 6 + 0], t01 = theta[n * 6 + 1], t02 = theta[n * 6 + 2];
    const float t10 = theta[n * 6 + 3], t11 = theta[n * 6 + 4], t12 = theta[n * 6 + 5];

    const float yn = -1.0f + 2.0f * (float)h / (float)(OUT_H - 1);

    // --- A matrix (16x4 f32): row0 = (t00,t01,t02,0) -> coords_x
    //                          row1 = (t10,t11,t12,0) -> coords_y
    // 32-bit A layout: VGPR0 = K0 (lanes 0-15) | K2 (lanes 16-31)
    //                  VGPR1 = K1 (lanes 0-15) | K3 (lanes 16-31)
    v2f A;
    A.x = (lane == 0) ? t00 : (lane == 1) ? t10 :
          (lane == 16) ? t02 : (lane == 17) ? t12 : 0.0f;
    A.y = (lane == 0) ? t01 : (lane == 1) ? t11 : 0.0f;

    // --- B matrix (4x16 f32): column j = (xn[j], yn, 1, 0)
    // 32-bit B layout (symmetric to A): VGPR0 = K0 | K2, VGPR1 = K1 | K3
    const float jf = (float)(jbase + (lane & 15));
    const float xn = -1.0f + 2.0f * jf / (wf - 1.0f);
    v2f B;
    B.x = (lane < 16) ? xn : 1.0f;
    B.y = (lane < 16) ? yn : 0.0f;

    v8f acc = {};
    // 8 args: (neg_a, A, neg_b, B, c_mod, C, reuse_a, reuse_b)
    acc = __builtin_amdgcn_wmma_f32_16x16x4_f32(
        false, A, false, B, (short)0, acc, false, false);

    // Lanes 0..15: acc[0] = coords_x[j], acc[1] = coords_y[j]  (j = jbase+lane)
    const float xs = 0.5f * (acc[0] + 1.0f) * (float)(FM_W - 1);
    const float ys = 0.5f * (acc[1] + 1.0f) * (float)(FM_H - 1);

    const float* fb   = fmap + (size_t)b * (FM_H * FM_W * FM_C);
    float*       orow = out + ((size_t)(n * OUT_H + h) * MAX_W + jbase) * FM_C;
    const int    c0   = lane * 8;                // 8 channels per lane (2 x B128)

    // Wave-uniform split: columns [0, nvalid) sample, [nvalid, TILE) are zero pad.
    int nvalid = wmax - jbase;
    nvalid = nvalid < 0 ? 0 : (nvalid > TILE ? TILE : nvalid);

    #pragma unroll 1
    for (int t = 0; t < nvalid; ++t) {
        // Uniform-lane broadcast of column t's source coords (v_readlane -> SGPR).
        const float x = readlane_f32(xs, t);
        const float y = readlane_f32(ys, t);
        float* op = orow + (size_t)t * FM_C + c0;

        int x0 = (int)floorf(x); x0 = x0 < 0 ? 0 : (x0 > FM_W - 1 ? FM_W - 1 : x0);
        int x1 = x0 + 1;         x1 = x1 > FM_W - 1 ? FM_W - 1 : x1;
        int y0 = (int)floorf(y); y0 = y0 < 0 ? 0 : (y0 > FM_H - 1 ? FM_H - 1 : y0);
        int y1 = y0 + 1;         y1 = y1 > FM_H - 1 ? FM_H - 1 : y1;

        const float x0f = (float)x0, x1f = (float)x1;
        const float y0f = (float)y0, y1f = (float)y1;
        const float wa = (x1f - x) * (y1f - y);
        const float wb = (x1f - x) * (y - y0f);
        const float wc = (x - x0f) * (y1f - y);
        const float wd = (x - x0f) * (y - y0f);

        const float* pa = fb + ((size_t)y0 * FM_W + x0) * FM_C + c0;
        const float* pb = fb + ((size_t)y1 * FM_W + x0) * FM_C + c0;
        const float* pc = fb + ((size_t)y0 * FM_W + x1) * FM_C + c0;
        const float* pd = fb + ((size_t)y1 * FM_W + x1) * FM_C + c0;

        const v4f a0  = *(const v4f*)(pa), a1  = *(const v4f*)(pa + 4);
        const v4f b0  = *(const v4f*)(pb), b1  = *(const v4f*)(pb + 4);
        const v4f cc0 = *(const v4f*)(pc), cc1 = *(const v4f*)(pc + 4);
        const v4f d0  = *(const v4f*)(pd), d1  = *(const v4f*)(pd + 4);

        const v4f r0 = wa * a0 + wb * b0 + wc * cc0 + wd * d0;
        const v4f r1 = wa * a1 + wb * b1 + wc * cc1 + wd * d1;

        store_nt_f4(op,     r0);
        store_nt_f4(op + 4, r1);
    }

    const v4f z = {0.0f, 0.0f, 0.0f, 0.0f};
    #pragma unroll 1
    for (int t = nvalid; t < TILE; ++t) {
        float* op = orow + (size_t)t * FM_C + c0;
        store_nt_f4(op,     z);
        store_nt_f4(op + 4, z);
    }
}

extern "C" void kernel_launch(void* const* d_in, const int* in_sizes, int n_in,
                              void* d_out, int out_size, void* d_ws, size_t ws_size,
                              hipStream_t stream) {
    (void)in_sizes; (void)n_in; (void)out_size; (void)d_ws; (void)ws_size;
    const float* feature_map = (const float*)d_in[0];
    const float* theta       = (const float*)d_in[1];
    const int*   box_masks   = (const int*)d_in[2];
    const int*   box_widths  = (const int*)d_in[3];
    float*       out         = (float*)d_out;

    // 1920 waves -> 480 blocks x 128 threads (4 wave32 per block)
    const int threads = 128;
    const int blocks  = (N_WAVES * 32) / threads;
    stn_warp_wmma_kernel<<<blocks, threads, 0, stream>>>(
        feature_map, theta, box_masks, box_widths, out);
}